// Model_38165079392620
// MI455X (gfx1250) — compile-verified
//
#include <hip/hip_runtime.h>
#include <hip/hip_bf16.h>

typedef __attribute__((ext_vector_type(2))) float v2f;
typedef __attribute__((ext_vector_type(8))) float v8f;
typedef __attribute__((ext_vector_type(4))) unsigned int u32x4;
typedef __attribute__((ext_vector_type(8))) int i32x8;
typedef __attribute__((ext_vector_type(4))) int i32x4;

#define KPTS 25
#define KPAD 28
#define HN 10
#define COUT 64
#define GBASE 128
#define NFLAGS (GBASE*GBASE*GBASE)          // 2,097,152
#define SCAN_ITEMS 8
#define SCAN_THREADS 256
#define SCAN_CHUNK (SCAN_ITEMS*SCAN_THREADS) // 2048
#define SCAN_BLOCKS (NFLAGS/SCAN_CHUNK)      // 1024

__device__ __forceinline__ int lin_of(float px, float py, float pz) {
    const float CELL = 0.1f;                 // == f32(0.05*2), matches jnp f32 division
    int gx = (int)floorf(px / CELL);
    int gy = (int)floorf(py / CELL);
    int gz = (int)floorf(pz / CELL);
    return (gx * GBASE + gy) * GBASE + gz;
}

// ---------------- init: zero flags, psum, counts, output features ----------------
__global__ void k_init(int* flags, float* psum, float* cnt, float* outF, int n) {
    int stride = gridDim.x * blockDim.x;
    int t0 = blockIdx.x * blockDim.x + threadIdx.x;
    for (int i = t0; i < NFLAGS;   i += stride) flags[i] = 0;
    for (int i = t0; i < 3 * n;    i += stride) psum[i]  = 0.0f;
    for (int i = t0; i < n;        i += stride) cnt[i]   = 0.0f;
    for (int i = t0; i < COUT * n; i += stride) outF[i]  = 0.0f;
}

// ---------------- mark occupied voxels ----------------
__global__ void k_flag(const float* __restrict__ pts, int* flags, int n) {
    int i = blockIdx.x * blockDim.x + threadIdx.x;
    if (i < n) {
        flags[lin_of(pts[3*i+0], pts[3*i+1], pts[3*i+2])] = 1;
    }
}

// ---------------- hierarchical exclusive scan over 2M flags ----------------
__global__ void k_scan1(const int* __restrict__ flags, int* __restrict__ bsum) {
    __shared__ int lds[SCAN_THREADS];
    int base = blockIdx.x * SCAN_CHUNK + threadIdx.x * SCAN_ITEMS;
    int s = 0;
    #pragma unroll
    for (int j = 0; j < SCAN_ITEMS; ++j) s += flags[base + j];
    lds[threadIdx.x] = s;
    __syncthreads();
    for (int off = SCAN_THREADS / 2; off > 0; off >>= 1) {
        if (threadIdx.x < off) lds[threadIdx.x] += lds[threadIdx.x + off];
        __syncthreads();
    }
    if (threadIdx.x == 0) bsum[blockIdx.x] = lds[0];
}

__global__ void k_scan2(int* bsum) {   // launched as 1 block x 1024 threads
    __shared__ int lds[SCAN_BLOCKS];
    int tid = threadIdx.x;
    int v = bsum[tid];
    lds[tid] = v;
    __syncthreads();
    for (int off = 1; off < SCAN_BLOCKS; off <<= 1) {
        int x = (tid >= off) ? lds[tid - off] : 0;
        __syncthreads();
        lds[tid] += x;
        __syncthreads();
    }
    bsum[tid] = lds[tid] - v;          // exclusive
}

__global__ void k_scan3(int* flags, const int* __restrict__ bsum) {
    __shared__ int lds[SCAN_THREADS];
    int tid = threadIdx.x;
    int base = blockIdx.x * SCAN_CHUNK + tid * SCAN_ITEMS;
    int v[SCAN_ITEMS];
    int s = 0;
    #pragma unroll
    for (int j = 0; j < SCAN_ITEMS; ++j) { v[j] = flags[base + j]; s += v[j]; }
    lds[tid] = s;
    __syncthreads();
    for (int off = 1; off < SCAN_THREADS; off <<= 1) {
        int x = (tid >= off) ? lds[tid - off] : 0;
        __syncthreads();
        lds[tid] += x;
        __syncthreads();
    }
    int run = bsum[blockIdx.x] + lds[tid] - s;   // exclusive prefix at this thread
    #pragma unroll
    for (int j = 0; j < SCAN_ITEMS; ++j) { int f = v[j]; flags[base + j] = run; run += f; }
}

// ---------------- fused KPConv + WMMA projection + pooled scatter ----------------
__global__ void __launch_bounds__(256) k_kpconv_wmma(
    const float* __restrict__ pts, const float* __restrict__ feats,
    const int* __restrict__ nidx, const float* __restrict__ kpts,
    const float* __restrict__ W, const int* __restrict__ rank,
    float* psum, float* cnt, float* outF, int n)
{
    __shared__ float fkLDS[256 * KPAD];
    __shared__ float Wlds[KPAD * COUT];
    __shared__ float kpx[KPTS], kpy[KPTS], kpz[KPTS], kp2s[KPTS];
    __shared__ int   uLDS[256];

    const float RADIUS = (float)(2.1 * 0.05);
    int tid = threadIdx.x;

#if defined(__HIP_DEVICE_COMPILE__) && __has_builtin(__builtin_amdgcn_tensor_load_to_lds)
    // --- TDM: DMA W[25][64] f32 into LDS as a 64x28 tile (rows 25..27 OOB -> zero) ---
    if (tid < 32) {   // wave 0 issues one TENSOR_LOAD_TO_LDS (EXEC-independent, per-wave)
        unsigned long long ga = (unsigned long long)(uintptr_t)W;
        unsigned ldsa = (unsigned)(uintptr_t)(void*)&Wlds[0];   // low 32 bits = LDS offset
        u32x4 g0;
        g0[0] = 1u;                                       // count=1, user descriptor
        g0[1] = ldsa;                                     // lds_addr
        g0[2] = (unsigned)(ga & 0xffffffffu);             // global_addr[31:0]
        g0[3] = (unsigned)((ga >> 32) & 0x01ffffffu)      // global_addr[56:32]
              | (2u << 30);                               // type = 2 ("image")
        i32x8 g1;
        g1[0] = (int)(2u << 16);      // data_size = 4 bytes
        g1[1] = (int)(64u << 16);     // tensor_dim0 = 64   (bits [79:48])
        g1[2] = (int)(25u << 16);     // tensor_dim1 = 25   (bits [111:80])
        g1[3] = (int)(64u << 16);     // tile_dim0   = 64   (bits [127:112])
        g1[4] = 28;                   // tile_dim1   = 28   (bits [143:128])
        g1[5] = 64;                   // tensor_dim0_stride = 64 (bits [207:160])
        g1[6] = 0;
        g1[7] = 0;
        i32x4 gz = {0, 0, 0, 0};      // groups 2/3 unused (<=2D tensor)
  #if __clang_major__ >= 23
        i32x8 gz8 = {0, 0, 0, 0, 0, 0, 0, 0};
        __builtin_amdgcn_tensor_load_to_lds(g0, g1, gz, gz, gz8, 0);
  #else
        __builtin_amdgcn_tensor_load_to_lds(g0, g1, gz, gz, 0);
  #endif
        __builtin_amdgcn_s_wait_tensorcnt(0);
    }
#else
    for (int j = tid; j < KPAD * COUT; j += 256)
        Wlds[j] = (j < KPTS * COUT) ? W[j] : 0.0f;
#endif
    if (tid < KPTS) {
        float x = kpts[tid*3+0], y = kpts[tid*3+1], z = kpts[tid*3+2];
        kpx[tid] = x; kpy[tid] = y; kpz[tid] = z;
        kp2s[tid] = x*x + y*y + z*z;
    }
    __syncthreads();

    int i = blockIdx.x * 256 + tid;
    float fk[KPTS];
    #pragma unroll
    for (int k = 0; k < KPTS; ++k) fk[k] = 0.0f;

    if (i < n) {
        float px = pts[3*i+0], py = pts[3*i+1], pz = pts[3*i+2];
        for (int h = 0; h < HN; ++h) {
            int j = nidx[i*HN + h];
            float rx = pts[3*j+0] - px;
            float ry = pts[3*j+1] - py;
            float rz = pts[3*j+2] - pz;
            float rel2 = rx*rx + ry*ry + rz*rz;
            float nf = feats[j];
            #pragma unroll
            for (int k = 0; k < KPTS; ++k) {
                float cross = rx*kpx[k] + ry*kpy[k] + rz*kpz[k];
                float d2   = fmaxf(rel2 + kp2s[k] - 2.0f*cross, 0.0f);
                float infl = fmaxf(1.0f - sqrtf(d2)/RADIUS, 0.0f);
                fk[k] += infl * nf;
            }
        }
        int u = rank[lin_of(px, py, pz)];
        uLDS[tid] = u;
        atomicAdd(&psum[u*3+0], px);
        atomicAdd(&psum[u*3+1], py);
        atomicAdd(&psum[u*3+2], pz);
        atomicAdd(&cnt[u], 1.0f);
    } else {
        uLDS[tid] = 0;
    }

    #pragma unroll
    for (int k = 0; k < KPTS; ++k) fkLDS[tid*KPAD + k] = fk[k];
    fkLDS[tid*KPAD + 25] = 0.0f;
    fkLDS[tid*KPAD + 26] = 0.0f;
    fkLDS[tid*KPAD + 27] = 0.0f;
    __syncthreads();

    // per-wave WMMA: fk[16x28] @ W[28x64], 7 K-steps of f32 16x16x4
    int w    = tid >> 5;
    int lane = tid & 31;
    int half = lane >> 4;     // 0: lanes 0-15 (K=0,1) ; 1: lanes 16-31 (K=2,3)
    int lm   = lane & 15;

    for (int mt = 0; mt < 2; ++mt) {
        v8f acc[4] = {};
        int arow = w*32 + mt*16 + lm;       // A row (M) held by this lane
        #pragma unroll
        for (int kk = 0; kk < 7; ++kk) {
            int ka = kk*4 + half*2;
            v2f a = *(const v2f*)&fkLDS[arow*KPAD + ka];   // A[m][ka], A[m][ka+1]
            #pragma unroll
            for (int nt = 0; nt < 4; ++nt) {
                int col = nt*16 + lm;
                v2f b;
                b[0] = Wlds[(ka+0)*COUT + col];            // B row ka   , col n
                b[1] = Wlds[(ka+1)*COUT + col];            // B row ka+1 , col n
                acc[nt] = __builtin_amdgcn_wmma_f32_16x16x4_f32(
                    false, a, false, b, (short)0, acc[nt], false, false);
            }
        }
        // D layout: VGPR r -> M=r (lanes 0-15) / M=r+8 (lanes 16-31), N=lane%16
        #pragma unroll
        for (int nt = 0; nt < 4; ++nt) {
            int col = nt*16 + lm;
            #pragma unroll
            for (int r = 0; r < 8; ++r) {
                int pml = w*32 + mt*16 + r + 8*half;
                int pm  = blockIdx.x*256 + pml;
                if (pm < n) {
                    float val = acc[nt][r];
                    if (val > 0.0f) {   // positive floats order like ints; 0-init covers clamp
                        atomicMax((int*)&outF[(size_t)uLDS[pml]*COUT + col],
                                  __float_as_int(val));
                    }
                }
            }
        }
    }
}

// ---------------- pooled point positions ----------------
__global__ void k_finalize(const float* __restrict__ psum, const float* __restrict__ cnt,
                           float* outP, int n) {
    int u = blockIdx.x * blockDim.x + threadIdx.x;
    if (u < n) {
        float c = fmaxf(cnt[u], 1.0f);
        outP[3*u+0] = psum[3*u+0] / c;
        outP[3*u+1] = psum[3*u+1] / c;
        outP[3*u+2] = psum[3*u+2] / c;
    }
}

extern "C" void kernel_launch(void* const* d_in, const int* in_sizes, int n_in,
                              void* d_out, int out_size, void* d_ws, size_t ws_size,
                              hipStream_t stream) {
    const float* pts   = (const float*)d_in[0];
    const float* feats = (const float*)d_in[1];
    const int*   nidx  = (const int*)d_in[2];
    const float* kpts  = (const float*)d_in[3];
    const float* W     = (const float*)d_in[4];
    int n = in_sizes[1];                     // features is [N, 1]

    char*  ws    = (char*)d_ws;
    int*   flags = (int*)ws;                                          // 8 MB
    int*   bsum  = (int*)(ws + (size_t)NFLAGS*4);                     // 4 KB
    float* psum  = (float*)(ws + (size_t)NFLAGS*4 + 4096);            // 3n floats
    float* cnt   = (float*)(ws + (size_t)NFLAGS*4 + 4096 + (size_t)3*n*4);
    float* outP  = (float*)d_out;
    float* outF  = (float*)d_out + (size_t)3*n;

    int nb = (n + 255) / 256;
    k_init<<<1024, 256, 0, stream>>>(flags, psum, cnt, outF, n);
    k_flag<<<nb, 256, 0, stream>>>(pts, flags, n);
    k_scan1<<<SCAN_BLOCKS, SCAN_THREADS, 0, stream>>>(flags, bsum);
    k_scan2<<<1, SCAN_BLOCKS, 0, stream>>>(bsum);
    k_scan3<<<SCAN_BLOCKS, SCAN_THREADS, 0, stream>>>(flags, bsum);
    k_kpconv_wmma<<<nb, 256, 0, stream>>>(pts, feats, nidx, kpts, W,
                                          flags, psum, cnt, outF, n);
    k_finalize<<<nb, 256, 0, stream>>>(psum, cnt, outP, n);
}